// RefCondMulConv_65472481460822
// MI455X (gfx1250) — compile-verified
//
#include <hip/hip_runtime.h>

typedef float v2f __attribute__((ext_vector_type(2)));
typedef float v8f __attribute__((ext_vector_type(8)));

#define CLASSES 128
#define K_IN    64
#define NOUT    32
#define NSAMP   65536
#define LDSPAD  68   // padded LDS row stride in floats (64 would be 16-way bank conflict)

// ---------------- bucketing kernels ----------------

__global__ void zero_ws_k(int* __restrict__ counts, int* __restrict__ cursors) {
    int i = threadIdx.x;
    if (i < CLASSES) { counts[i] = 0; cursors[i] = 0; }
}

__global__ void hist_k(const long long* __restrict__ inds, int* __restrict__ counts) {
    int i = blockIdx.x * blockDim.x + threadIdx.x;
    if (i < NSAMP) atomicAdd(&counts[(int)inds[i]], 1);
}

__global__ void scan_k(const int* __restrict__ counts,
                       int* __restrict__ offsets, int* __restrict__ tileOff) {
    if (threadIdx.x == 0) {
        int s = 0, t = 0;
        for (int c = 0; c < CLASSES; ++c) {
            offsets[c] = s; tileOff[c] = t;
            s += counts[c];
            t += (counts[c] + 15) >> 4;
        }
        offsets[CLASSES] = s; tileOff[CLASSES] = t;
    }
}

__global__ void scatter_k(const long long* __restrict__ inds,
                          const int* __restrict__ offsets,
                          int* __restrict__ cursors, int* __restrict__ bucket) {
    int i = blockIdx.x * blockDim.x + threadIdx.x;
    if (i < NSAMP) {
        int c = (int)inds[i];
        int p = atomicAdd(&cursors[c], 1);
        bucket[offsets[c] + p] = i;
    }
}

// ---------------- grouped gather-GEMM with fp32 WMMA ----------------
// One wave32 block per 16-sample tile of one class:
//   D(16x32) = Xg(16x64) @ Wc^T(64x32) + bias   via  2 x 16 v_wmma_f32_16x16x4_f32

__global__ __launch_bounds__(32) void cond_gemm_wmma_k(
    const float* __restrict__ x, const float* __restrict__ W,
    const float* __restrict__ bias,
    const int* __restrict__ offsets, const int* __restrict__ tileOff,
    const int* __restrict__ bucket, float* __restrict__ out)
{
    __shared__ float lsA[16 * LDSPAD];   // gathered X rows   (16 x 64, padded)
    __shared__ float lsW[32 * LDSPAD];   // W[c] slice        (32 x 64, padded)
    __shared__ int   lsId[16];

    const int b = blockIdx.x;
    const int total = tileOff[CLASSES];
    if (b >= total) return;                       // uniform per block

    // binary search: largest c with tileOff[c] <= b
    int lo = 0, hi = CLASSES - 1;
    while (lo < hi) {
        int mid = (lo + hi + 1) >> 1;
        if (tileOff[mid] <= b) lo = mid; else hi = mid - 1;
    }
    const int c    = lo;
    const int base = offsets[c] + 16 * (b - tileOff[c]);
    int cnt = offsets[c + 1] - base; if (cnt > 16) cnt = 16;

    const int t = threadIdx.x;

    // stage gathered X rows: two threads per row, 8 float4 each
    {
        const int row = t >> 1, half = t & 1;
        const int sid = (row < cnt) ? bucket[base + row] : -1;
        if (half == 0) lsId[row] = sid;
        const float4* src =
            (const float4*)(x + (size_t)(sid < 0 ? 0 : sid) * K_IN + half * 32);
        #pragma unroll
        for (int q = 0; q < 8; ++q) {
            float4 v = (sid >= 0) ? src[q] : make_float4(0.f, 0.f, 0.f, 0.f);
            float* dst = &lsA[row * LDSPAD + half * 32 + q * 4];
            dst[0] = v.x; dst[1] = v.y; dst[2] = v.z; dst[3] = v.w;
        }
    }
    // stage W[c] (32x64 contiguous): one row per thread, 16 float4 each
    {
        const float4* src = (const float4*)(W + ((size_t)c * NOUT + t) * K_IN);
        #pragma unroll
        for (int q = 0; q < 16; ++q) {
            float4 v = src[q];
            float* dst = &lsW[t * LDSPAD + q * 4];
            dst[0] = v.x; dst[1] = v.y; dst[2] = v.z; dst[3] = v.w;
        }
    }
    __syncthreads();

    // WMMA f32 16x16x4 layout: lanes 0-15 -> (K0,K1), lanes 16-31 -> (K2,K3)
    const int col = t & 15;
    const int klo = (t >> 4) << 1;
    v8f acc0 = {0,0,0,0,0,0,0,0};
    v8f acc1 = {0,0,0,0,0,0,0,0};

    #pragma unroll
    for (int kk = 0; kk < 16; ++kk) {
        const int k0 = kk * 4 + klo;
        v2f a  = { lsA[col * LDSPAD + k0],        lsA[col * LDSPAD + k0 + 1] };
        v2f b0 = { lsW[col * LDSPAD + k0],        lsW[col * LDSPAD + k0 + 1] };
        v2f b1 = { lsW[(col + 16) * LDSPAD + k0], lsW[(col + 16) * LDSPAD + k0 + 1] };
        acc0 = __builtin_amdgcn_wmma_f32_16x16x4_f32(
            false, a, false, b0, (short)0, acc0, false, false);
        acc1 = __builtin_amdgcn_wmma_f32_16x16x4_f32(
            false, a, false, b1, (short)0, acc1, false, false);
    }

    // epilogue: fused bias + scatter to out[sid, :]
    const float bv0 = bias[c * NOUT + col];
    const float bv1 = bias[c * NOUT + 16 + col];
    const int mHi = (t >> 4) * 8;
    #pragma unroll
    for (int v = 0; v < 8; ++v) {
        const int m = v + mHi;           // D VGPR v: M=v (lanes 0-15), M=v+8 (16-31)
        if (m < cnt) {
            const int sid = lsId[m];
            out[(size_t)sid * NOUT + col]      = acc0[v] + bv0;
            out[(size_t)sid * NOUT + 16 + col] = acc1[v] + bv1;
        }
    }
}

// ---------------- launcher ----------------

extern "C" void kernel_launch(void* const* d_in, const int* in_sizes, int n_in,
                              void* d_out, int out_size, void* d_ws, size_t ws_size,
                              hipStream_t stream) {
    const float*     x    = (const float*)d_in[0];
    const long long* inds = (const long long*)d_in[1];
    const float*     W    = (const float*)d_in[2];
    const float*     bias = (const float*)d_in[3];
    float*           out  = (float*)d_out;

    int* counts  = (int*)d_ws;            // 128
    int* offsets = counts  + CLASSES;     // 129
    int* tileOff = offsets + CLASSES + 1; // 129
    int* cursors = tileOff + CLASSES + 1; // 128
    int* bucket  = cursors + CLASSES;     // 65536

    zero_ws_k<<<1, 128, 0, stream>>>(counts, cursors);
    hist_k<<<NSAMP / 256, 256, 0, stream>>>(inds, counts);
    scan_k<<<1, 32, 0, stream>>>(counts, offsets, tileOff);
    scatter_k<<<NSAMP / 256, 256, 0, stream>>>(inds, offsets, cursors, bucket);

    // max tiles = 65536/16 + 128 = 4224 (blocks past the real total exit early)
    cond_gemm_wmma_k<<<4224, 32, 0, stream>>>(x, W, bias, offsets, tileOff, bucket, out);
}